// LDGNNetwork_59803124630338
// MI455X (gfx1250) — compile-verified
//
#include <hip/hip_runtime.h>
#include <hip/hip_bf16.h>
#include <cmath>

typedef __attribute__((ext_vector_type(2))) float v2f;
typedef __attribute__((ext_vector_type(8))) float v8f;

#define BSZ 128
#define NNODE 32
#define IN_DIM 16
#define HID 128
#define HEADS 4
#define HD 512          // HEADS*HID
#define LATENT 1152
#define DUEL_H 256
#define NOUT 5

#define AS_STRIDE 34    // padded LDS row stride (words): 8B-aligned, conflict-free

// ---------------------------------------------------------------------------
// prep: extract per-node features and adjacency bias (0 / -inf) from obs
// ---------------------------------------------------------------------------
__global__ void __launch_bounds__(256) prep_kernel(const float* __restrict__ obs,
                                                   float* __restrict__ FEATS,
                                                   float* __restrict__ ADJB) {
  const int b = blockIdx.x;
  const int tid = threadIdx.x;
  const float* o = obs + (size_t)b * 577;
  for (int p = tid; p < NNODE * IN_DIM; p += 256) {
    int i = p >> 4, c = p & 15;
    FEATS[(size_t)b * (NNODE * IN_DIM) + p] = o[i * 18 + 2 + c];
  }
  for (int p = tid; p < NNODE * NNODE; p += 256) {
    int i = p >> 5, j = p & 31;
    float dx = o[i * 18 + 0] - o[j * 18 + 0];
    float dy = o[i * 18 + 1] - o[j * 18 + 1];
    float d2 = dx * dx + dy * dy;
    ADJB[(size_t)b * (NNODE * NNODE) + p] = (d2 < 0.09f || i == j) ? 0.0f : -INFINITY;
  }
}

// ---------------------------------------------------------------------------
// fp32 WMMA GEMM:  C[M,N] = act(A[M,K] @ B[K,N] + bias)
// Block = 256 threads (8 waves) computes a 128M x 128N macro-tile.
// grid = (M/128, N/128). Wave w owns N-stripe w and 8 M-tile accumulators:
// per 4-wide K step it loads one B pair and issues 8 WMMAs (8x B reuse).
// A staged through LDS in 32-wide K chunks (padded stride, b64-aligned).
// ---------------------------------------------------------------------------
__global__ void __launch_bounds__(256) gemm_wmma_f32(const float* __restrict__ A,
                                                     const float* __restrict__ B,
                                                     const float* __restrict__ bias,
                                                     float* __restrict__ C,
                                                     int M, int N, int K, int relu) {
  __shared__ float As[128 * AS_STRIDE];        // ~17 KB
  const int tid  = threadIdx.x;
  const int wave = tid >> 5;
  const int lane = tid & 31;
  const int l15  = lane & 15;
  const int koff = (lane >> 4) << 1;           // 0 for lanes 0-15, 2 for 16-31
  const int m0     = blockIdx.x * 128;
  const int n_tile = blockIdx.y * 128 + wave * 16;

  v8f acc[8];
#pragma unroll
  for (int mt = 0; mt < 8; ++mt) acc[mt] = (v8f){};

  const float* Bp = B + n_tile + l15;

  for (int k0 = 0; k0 < K; k0 += 32) {
    int Kc = K - k0; if (Kc > 32) Kc = 32;
    __syncthreads();
    for (int idx = tid; idx < 128 * Kc; idx += 256) {
      int r = idx / Kc, c = idx - r * Kc;
      As[r * AS_STRIDE + c] = A[(size_t)(m0 + r) * K + k0 + c];
    }
    __syncthreads();

    if (k0 + 32 < K)   // prefetch next B chunk (global_prefetch_b8)
      __builtin_prefetch(&Bp[(size_t)(k0 + 32 + koff) * N], 0, 3);

    if (Kc == 32) {
#pragma unroll
      for (int ks = 0; ks < 32; ks += 4) {
        v2f bb;
        bb.x = Bp[(size_t)(k0 + ks + koff) * N];
        bb.y = Bp[(size_t)(k0 + ks + koff + 1) * N];
#pragma unroll
        for (int mt = 0; mt < 8; ++mt) {
          v2f a;
          a.x = As[(mt * 16 + l15) * AS_STRIDE + ks + koff];
          a.y = As[(mt * 16 + l15) * AS_STRIDE + ks + koff + 1];
          acc[mt] = __builtin_amdgcn_wmma_f32_16x16x4_f32(false, a, false, bb,
                                                          (short)0, acc[mt], false, false);
        }
      }
    } else {
      for (int ks = 0; ks < Kc; ks += 4) {
        v2f bb;
        bb.x = Bp[(size_t)(k0 + ks + koff) * N];
        bb.y = Bp[(size_t)(k0 + ks + koff + 1) * N];
#pragma unroll
        for (int mt = 0; mt < 8; ++mt) {
          v2f a;
          a.x = As[(mt * 16 + l15) * AS_STRIDE + ks + koff];
          a.y = As[(mt * 16 + l15) * AS_STRIDE + ks + koff + 1];
          acc[mt] = __builtin_amdgcn_wmma_f32_16x16x4_f32(false, a, false, bb,
                                                          (short)0, acc[mt], false, false);
        }
      }
    }
  }

  const int col = n_tile + l15;
  const float bv = bias ? bias[col] : 0.0f;
  const int rb = (lane >> 4) << 3;
#pragma unroll
  for (int mt = 0; mt < 8; ++mt) {
    const int rbase = m0 + mt * 16 + rb;
#pragma unroll
    for (int v = 0; v < 8; ++v) {
      float val = acc[mt][v] + bv;
      if (relu) val = fmaxf(val, 0.0f);
      C[(size_t)(rbase + v) * N + col] = val;
    }
  }
}

// ---------------------------------------------------------------------------
// GATv2 attention: one block per graph; loop over heads.
//  logits_ij = sum_c lrelu(gr_i[c]+gl_j[c])*att[c]  (+ adj bias)
//  alpha = softmax_j ; out = relu(alpha @ gl + bias)   (aggregation via WMMA)
// ---------------------------------------------------------------------------
__global__ void __launch_bounds__(256) gat_attn(const float* __restrict__ GL,
                                                const float* __restrict__ GR,
                                                const float* __restrict__ ADJB,
                                                const float* __restrict__ att,
                                                const float* __restrict__ bias,
                                                float* __restrict__ XG) {
  const int b = blockIdx.x;
  const int tid = threadIdx.x;
  const int wave = tid >> 5;
  const int lane = tid & 31;

  __shared__ float s_gl[NNODE * HID];     // 16 KB
  __shared__ float s_gr[NNODE * HID];     // 16 KB
  __shared__ float s_att[HID];
  __shared__ float s_adj[NNODE * NNODE];  // 4 KB
  __shared__ float s_log[NNODE * 33];     // padded 32x32

  for (int p = tid; p < NNODE * NNODE; p += 256)
    s_adj[p] = ADJB[(size_t)b * (NNODE * NNODE) + p];

  const size_t gbase = (size_t)b * NNODE * HD;

  for (int h = 0; h < HEADS; ++h) {
    __syncthreads();
    for (int p = tid; p < NNODE * HID; p += 256) {
      int i = p >> 7, c = p & 127;
      s_gl[p] = GL[gbase + (size_t)i * HD + h * HID + c];
      s_gr[p] = GR[gbase + (size_t)i * HD + h * HID + c];
    }
    if (tid < HID) s_att[tid] = att[h * HID + tid];
    __syncthreads();

    // logits
    for (int p = tid; p < NNODE * NNODE; p += 256) {
      int i = p >> 5, j = p & 31;
      const float* gr = &s_gr[i * HID];
      const float* gl = &s_gl[j * HID];
      float acc = 0.0f;
#pragma unroll 8
      for (int c = 0; c < HID; ++c) {
        float t = gr[c] + gl[c];
        t = (t > 0.0f) ? t : 0.2f * t;
        acc += t * s_att[c];
      }
      s_log[i * 33 + j] = acc + s_adj[p];
    }
    __syncthreads();

    // per-row softmax over j (rows always have the diagonal => finite max)
    if (tid < NNODE) {
      const int i = tid;
      float mx = -INFINITY;
      for (int j = 0; j < NNODE; ++j) mx = fmaxf(mx, s_log[i * 33 + j]);
      float sum = 0.0f;
      for (int j = 0; j < NNODE; ++j) {
        float e = __expf(s_log[i * 33 + j] - mx);
        s_log[i * 33 + j] = e;
        sum += e;
      }
      float inv = 1.0f / sum;
      for (int j = 0; j < NNODE; ++j) s_log[i * 33 + j] *= inv;
    }
    __syncthreads();

    // aggregation: out(32x128) = alpha(32x32) @ gl(32x128), via WMMA
    const int l15  = lane & 15;
    const int koff = (lane >> 4) << 1;
    for (int t = wave; t < 16; t += 8) {      // 2 Mtiles x 8 Ntiles
      int mt = t >> 3, nt = t & 7;
      v8f acc = {};
#pragma unroll
      for (int ks = 0; ks < NNODE; ks += 4) {
        v2f a, bb;
        a.x  = s_log[(mt * 16 + l15) * 33 + ks + koff];
        a.y  = s_log[(mt * 16 + l15) * 33 + ks + koff + 1];
        bb.x = s_gl[(ks + koff) * HID + nt * 16 + l15];
        bb.y = s_gl[(ks + koff + 1) * HID + nt * 16 + l15];
        acc = __builtin_amdgcn_wmma_f32_16x16x4_f32(false, a, false, bb,
                                                    (short)0, acc, false, false);
      }
      const int col = nt * 16 + l15;
      const float bv = bias[h * HID + col];
      const int rbase = mt * 16 + ((lane >> 4) << 3);
#pragma unroll
      for (int v = 0; v < 8; ++v) {
        float val = fmaxf(acc[v] + bv, 0.0f);
        XG[gbase + (size_t)(rbase + v) * HD + h * HID + col] = val;
      }
    }
  }
}

// ---------------------------------------------------------------------------
// gather x1/x2/x3 at agent index into x_cat (LATENT = 128+512+512)
// ---------------------------------------------------------------------------
__global__ void __launch_bounds__(128) gather_kernel(const float* __restrict__ obs,
                                                     const float* __restrict__ X,
                                                     const float* __restrict__ XG1,
                                                     const float* __restrict__ XG2,
                                                     float* __restrict__ XCAT) {
  const int b = blockIdx.x;
  const int tid = threadIdx.x;
  float af = obs[(size_t)b * 577 + 576];
  int ai = (int)fminf(fmaxf(af, 0.0f), 31.0f);
  float* dst = XCAT + (size_t)b * LATENT;
  const float* s1 = X   + (size_t)b * NNODE * HID + (size_t)ai * HID;
  const float* s2 = XG1 + (size_t)b * NNODE * HD  + (size_t)ai * HD;
  const float* s3 = XG2 + (size_t)b * NNODE * HD  + (size_t)ai * HD;
  for (int t = tid; t < LATENT; t += 128) {
    float v;
    if (t < HID)            v = s1[t];
    else if (t < HID + HD)  v = s2[t - HID];
    else                    v = s3[t - HID - HD];
    dst[t] = v;
  }
}

// ---------------------------------------------------------------------------
// final: q = qh @ q_w2 + b ; v = vh @ v_w2 + b ; out = q - mean(q) + v
// ---------------------------------------------------------------------------
__global__ void __launch_bounds__(64) final_kernel(const float* __restrict__ QH,
                                                   const float* __restrict__ VH,
                                                   const float* __restrict__ qw2,
                                                   const float* __restrict__ qb2,
                                                   const float* __restrict__ vw2,
                                                   const float* __restrict__ vb2,
                                                   float* __restrict__ out) {
  const int b = blockIdx.x;
  const int t = threadIdx.x;
  __shared__ float sq[NOUT];
  __shared__ float sv;
  const float* qh = QH + (size_t)b * DUEL_H;
  const float* vh = VH + (size_t)b * DUEL_H;
  if (t < NOUT) {
    float acc = qb2[t];
    for (int c = 0; c < DUEL_H; ++c) acc += qh[c] * qw2[c * NOUT + t];
    sq[t] = acc;
  } else if (t == NOUT) {
    float acc = vb2[0];
    for (int c = 0; c < DUEL_H; ++c) acc += vh[c] * vw2[c];
    sv = acc;
  }
  __syncthreads();
  if (t < NOUT) {
    float m = (sq[0] + sq[1] + sq[2] + sq[3] + sq[4]) * 0.2f;
    out[(size_t)b * NOUT + t] = sq[t] - m + sv;
  }
}

// ---------------------------------------------------------------------------
extern "C" void kernel_launch(void* const* d_in, const int* in_sizes, int n_in,
                              void* d_out, int out_size, void* d_ws, size_t ws_size,
                              hipStream_t stream) {
  (void)in_sizes; (void)n_in; (void)out_size; (void)ws_size;
  const float* obs    = (const float*)d_in[0];
  const float* enc_w1 = (const float*)d_in[1];
  const float* enc_b1 = (const float*)d_in[2];
  const float* enc_w2 = (const float*)d_in[3];
  const float* enc_b2 = (const float*)d_in[4];
  const float* wl1    = (const float*)d_in[5];
  const float* wr1    = (const float*)d_in[6];
  const float* att1   = (const float*)d_in[7];
  const float* bias1  = (const float*)d_in[8];
  const float* wl2    = (const float*)d_in[9];
  const float* wr2    = (const float*)d_in[10];
  const float* att2   = (const float*)d_in[11];
  const float* bias2  = (const float*)d_in[12];
  const float* q_w1   = (const float*)d_in[13];
  const float* q_b1   = (const float*)d_in[14];
  const float* q_w2   = (const float*)d_in[15];
  const float* q_b2   = (const float*)d_in[16];
  const float* v_w1   = (const float*)d_in[17];
  const float* v_b1   = (const float*)d_in[18];
  const float* v_w2   = (const float*)d_in[19];
  const float* v_b2   = (const float*)d_in[20];
  float* out = (float*)d_out;

  // workspace layout (floats)
  float* ws    = (float*)d_ws;
  float* FEATS = ws;                       // 4096*16
  float* H     = FEATS + 4096 * 16;        // 4096*128
  float* X     = H     + 4096 * 128;       // 4096*128
  float* ADJB  = X     + 4096 * 128;       // 128*32*32
  float* GL    = ADJB  + 128 * 1024;       // 4096*512 (reused for both layers)
  float* GR    = GL    + 4096 * 512;       // 4096*512 (reused)
  float* XG1   = GR    + 4096 * 512;       // 4096*512
  float* XG2   = XG1   + 4096 * 512;       // 4096*512
  float* XCAT  = XG2   + 4096 * 512;       // 128*1152
  float* QH    = XCAT  + 128 * LATENT;     // 128*256
  float* VH    = QH    + 128 * DUEL_H;     // 128*256

  const int M = BSZ * NNODE;               // 4096

  prep_kernel<<<BSZ, 256, 0, stream>>>(obs, FEATS, ADJB);

  // encoder
  gemm_wmma_f32<<<dim3(M / 128, 1), 256, 0, stream>>>(FEATS, enc_w1, enc_b1, H, M, HID, IN_DIM, 1);
  gemm_wmma_f32<<<dim3(M / 128, 1), 256, 0, stream>>>(H, enc_w2, enc_b2, X, M, HID, HID, 1);

  // GAT layer 1
  gemm_wmma_f32<<<dim3(M / 128, HD / 128), 256, 0, stream>>>(X, wl1, nullptr, GL, M, HD, HID, 0);
  gemm_wmma_f32<<<dim3(M / 128, HD / 128), 256, 0, stream>>>(X, wr1, nullptr, GR, M, HD, HID, 0);
  gat_attn<<<BSZ, 256, 0, stream>>>(GL, GR, ADJB, att1, bias1, XG1);

  // GAT layer 2
  gemm_wmma_f32<<<dim3(M / 128, HD / 128), 256, 0, stream>>>(XG1, wl2, nullptr, GL, M, HD, HD, 0);
  gemm_wmma_f32<<<dim3(M / 128, HD / 128), 256, 0, stream>>>(XG1, wr2, nullptr, GR, M, HD, HD, 0);
  gat_attn<<<BSZ, 256, 0, stream>>>(GL, GR, ADJB, att2, bias2, XG2);

  // heads
  gather_kernel<<<BSZ, 128, 0, stream>>>(obs, X, XG1, XG2, XCAT);
  gemm_wmma_f32<<<dim3(BSZ / 128, DUEL_H / 128), 256, 0, stream>>>(XCAT, q_w1, q_b1, QH, BSZ, DUEL_H, LATENT, 1);
  gemm_wmma_f32<<<dim3(BSZ / 128, DUEL_H / 128), 256, 0, stream>>>(XCAT, v_w1, v_b1, VH, BSZ, DUEL_H, LATENT, 1);
  final_kernel<<<BSZ, 64, 0, stream>>>(QH, VH, q_w2, q_b2, v_w2, v_b2, out);
}